// HierarchicalLLLoss_25967372272104
// MI455X (gfx1250) — compile-verified
//
#include <hip/hip_runtime.h>

// HierarchicalLLLoss for MI455X (gfx1250, wave32).
// B=512, C=2048, E=12.  num/den ratio is scale-invariant => use unnormalized
// exp(x - max) and skip the softmax normalization entirely.
// Contraction done with V_WMMA_F32_16X16X4_F32; B-matrix tiles double-buffered
// into LDS with global_load_async_to_lds_b128 (ASYNCcnt, in-order completion:
// s_wait_asynccnt 6 waits only for the previous tile's 6 loads).

typedef float v2f __attribute__((ext_vector_type(2)));
typedef float v8f __attribute__((ext_vector_type(8)));

#define B_SZ 512
#define C_SZ 2048
#define E_SZ 12
#define WAVES 8
#define TILE_ROWS 32                 // c-rows staged per async tile
#define KRANGE (C_SZ / WAVES)        // 256 c-rows per wave
#define NTILES (KRANGE / TILE_ROWS)  // 8 tiles per wave
#define TILE_F (TILE_ROWS * E_SZ)    // 384 floats = 1536 B per tile per matrix

__global__ void hll_zero_out(float* out) { out[0] = 0.0f; }

__device__ __forceinline__ void async_copy16(unsigned lds_addr, unsigned long long gaddr) {
  // GLOBAL_LOAD_ASYNC_TO_LDS_B128: vdst = LDS byte address VGPR, vaddr = 64-bit global addr
  asm volatile("global_load_async_to_lds_b128 %0, %1, off"
               :: "v"(lds_addr), "v"(gaddr) : "memory");
}
__device__ __forceinline__ void wait_async_le0() {
  asm volatile("s_wait_asynccnt 0" ::: "memory");
}
__device__ __forceinline__ void wait_async_le6() {
  // 6 loads per tile; async loads complete in order, so cnt<=6 means the
  // older (currently consumed) tile's loads have all landed in LDS.
  asm volatile("s_wait_asynccnt 6" ::: "memory");
}

// Issue one tile's worth of async copies (3 x B128 per lane per matrix).
__device__ __forceinline__ void issue_tile(unsigned ldsN, unsigned ldsD,
                                           unsigned long long srcN,
                                           unsigned long long srcD,
                                           int lane) {
  #pragma unroll
  for (int j = 0; j < (TILE_F * 4) / (32 * 16); ++j) {   // 3 chunks/lane
    unsigned off = (unsigned)(lane + j * 32) * 16u;
    async_copy16(ldsN + off, srcN + off);
    async_copy16(ldsD + off, srcD + off);
  }
}

__global__ void __launch_bounds__(256) hll_kernel(
    const float* __restrict__ inputs,   // [B, C]
    const int*   __restrict__ target,   // [B]
    const float* __restrict__ onum,     // [C, C, E]
    const float* __restrict__ oden,     // [C, C, E]
    const float* __restrict__ weights,  // [C, E]
    float* __restrict__ out)            // [1]
{
  __shared__ float probs[C_SZ];                  // 8 KB
  __shared__ float tileN[WAVES][2][TILE_F];      // 24 KB (double-buffered)
  __shared__ float tileD[WAVES][2][TILE_F];      // 24 KB
  __shared__ float red_num[WAVES][16];
  __shared__ float red_den[WAVES][16];
  __shared__ float smax[WAVES];

  const int b    = blockIdx.x;
  const int tid  = threadIdx.x;
  const int wave = tid >> 5;
  const int lane = tid & 31;
  const int t    = target[b];

  const float* row = inputs + (size_t)b * C_SZ;

  // ---- 1) row max (stability) ----
  float m = -INFINITY;
  for (int c = tid; c < C_SZ; c += 256) m = fmaxf(m, row[c]);
  #pragma unroll
  for (int s = 16; s >= 1; s >>= 1) m = fmaxf(m, __shfl_xor(m, s, 32));
  if (lane == 0) smax[wave] = m;
  __syncthreads();
  float gmax = smax[0];
  #pragma unroll
  for (int w = 1; w < WAVES; ++w) gmax = fmaxf(gmax, smax[w]);

  // ---- 2) unnormalized probs; num/den ratio is invariant to the softmax sum ----
  for (int c = tid; c < C_SZ; c += 256) probs[c] = __expf(row[c] - gmax);
  __syncthreads();

  // ---- 3) contraction via V_WMMA_F32_16X16X4_F32 (M-row 0 carries the data) ----
  const size_t slice = (size_t)t * (size_t)C_SZ * (size_t)E_SZ;
  const unsigned long long gN = (unsigned long long)(const void*)(onum + slice);
  const unsigned long long gD = (unsigned long long)(const void*)(oden + slice);

  v8f accN = {}; v8f accD = {};
  const int  m16    = lane & 15;
  const int  k0     = (lane >> 4) * 2;      // lanes 0-15: K=0,1 ; lanes 16-31: K=2,3
  const bool isRow0 = (m16 == 0);
  const bool nValid = (m16 < E_SZ);
  const int  nClamp = nValid ? m16 : (E_SZ - 1);   // branch-free B gather

  const unsigned ldsN0 = (unsigned)(unsigned long long)(void*)&tileN[wave][0][0];
  const unsigned ldsN1 = (unsigned)(unsigned long long)(void*)&tileN[wave][1][0];
  const unsigned ldsD0 = (unsigned)(unsigned long long)(void*)&tileD[wave][0][0];
  const unsigned ldsD1 = (unsigned)(unsigned long long)(void*)&tileD[wave][1][0];

  const int kbeg = wave * KRANGE;

  // Prologue: stage tile 0 into buffer 0.
  issue_tile(ldsN0, ldsD0,
             gN + (unsigned long long)kbeg * (E_SZ * 4),
             gD + (unsigned long long)kbeg * (E_SZ * 4), lane);

  for (int i = 0; i < NTILES; ++i) {
    const int cur = i & 1;
    if (i + 1 < NTILES) {
      const unsigned long long boff =
          (unsigned long long)(kbeg + (i + 1) * TILE_ROWS) * (E_SZ * 4);
      issue_tile(cur ? ldsN0 : ldsN1, cur ? ldsD0 : ldsD1,
                 gN + boff, gD + boff, lane);
      wait_async_le6();     // tile i's 6 loads complete (in-order)
    } else {
      wait_async_le0();     // last tile: drain everything
    }

    const float* tN = &tileN[wave][cur][0];
    const float* tD = &tileD[wave][cur][0];
    const int c0 = kbeg + i * TILE_ROWS;

    #pragma unroll
    for (int kk = 0; kk < TILE_ROWS; kk += 4) {
      // A fragment (16x4 f32): row 0 = probs chunk, rows 1..15 zero (select, no branch)
      const float p0 = probs[c0 + kk + k0 + 0];
      const float p1 = probs[c0 + kk + k0 + 1];
      v2f a; a.x = isRow0 ? p0 : 0.0f;
             a.y = isRow0 ? p1 : 0.0f;
      // B fragments (4x16 f32): B[k,n] = slice[(c0+kk+k)*E + n], n >= E -> 0
      const int r0 = (kk + k0) * E_SZ + nClamp;
      const float n0 = tN[r0], n1 = tN[r0 + E_SZ];
      const float d0 = tD[r0], d1 = tD[r0 + E_SZ];
      v2f bN; bN.x = nValid ? n0 : 0.0f;  bN.y = nValid ? n1 : 0.0f;
      v2f bD; bD.x = nValid ? d0 : 0.0f;  bD.y = nValid ? d1 : 0.0f;

      accN = __builtin_amdgcn_wmma_f32_16x16x4_f32(false, a, false, bN,
                                                   (short)0, accN, false, false);
      accD = __builtin_amdgcn_wmma_f32_16x16x4_f32(false, a, false, bD,
                                                   (short)0, accD, false, false);
    }
  }

  // ---- 4) D row 0 lives in acc[0], lanes 0..15 (N = lane). Reduce across waves ----
  if (lane < 16) { red_num[wave][lane] = accN[0]; red_den[wave][lane] = accD[0]; }
  __syncthreads();

  if (tid < 16) {
    float n = 0.0f, d = 0.0f;
    #pragma unroll
    for (int w = 0; w < WAVES; ++w) { n += red_num[w][tid]; d += red_den[w][tid]; }
    float v = 0.0f;
    if (tid < E_SZ && n != 0.0f) {
      const float wgt = weights[(size_t)t * E_SZ + tid];
      v = wgt * (-logf(n / d));
    }
    #pragma unroll
    for (int s = 8; s >= 1; s >>= 1) v += __shfl_xor(v, s, 32);
    if (tid == 0) atomicAdd(out, v * (1.0f / (float)B_SZ));
  }
}

extern "C" void kernel_launch(void* const* d_in, const int* in_sizes, int n_in,
                              void* d_out, int out_size, void* d_ws, size_t ws_size,
                              hipStream_t stream) {
  const float* inputs  = (const float*)d_in[0];
  const int*   target  = (const int*)  d_in[1];
  const float* onum    = (const float*)d_in[2];
  const float* oden    = (const float*)d_in[3];
  const float* weights = (const float*)d_in[4];
  float* out = (float*)d_out;

  hll_zero_out<<<1, 1, 0, stream>>>(out);
  hll_kernel<<<dim3(B_SZ), dim3(256), 0, stream>>>(inputs, target, onum, oden, weights, out);
}